// TrueHigherOrderAttention_41085657154100
// MI455X (gfx1250) — compile-verified
//
#include <hip/hip_runtime.h>
#include <math.h>

// Problem constants (match reference)
#define TT  512
#define CC  768
#define NHH 12
#define HSS 64
constexpr float SCALE = 0.125f; // 1/sqrt(64)

typedef float v2f __attribute__((ext_vector_type(2)));
typedef float v8f __attribute__((ext_vector_type(8)));

// CDNA5 full-precision WMMA: D(16x16 f32) = A(16x4 f32) * B(4x16 f32) + C
static __device__ __forceinline__ v8f wmma4(v2f a, v2f b, v8f c) {
  return __builtin_amdgcn_wmma_f32_16x16x4_f32(
      /*neg_a=*/false, a, /*neg_b=*/false, b,
      /*c_mod=*/(short)0, c, /*reuse_a=*/false, /*reuse_b=*/false);
}

// ---------------------------------------------------------------------------
// Kernel 1: fused QKV projections. P[w][h][t][d] = (x @ W_w)[t][h*64+d]
// One wave per 16x16 output tile; K = 768 in steps of 4 (192 v_wmma per tile).
// ---------------------------------------------------------------------------
__global__ void qkv_proj_kernel(const float* __restrict__ x,
                                const float* __restrict__ W0,
                                const float* __restrict__ W1,
                                const float* __restrict__ W2,
                                float* __restrict__ P) {
  const int lane = threadIdx.x;
  const int lm   = lane & 15;         // row/col within tile
  const int lk   = (lane >> 4) << 1;  // K sub-offset (0 or 2)
  const int drow = (lane >> 4) << 3;  // D-row offset (0 or 8)
  const int t0 = blockIdx.x * 16;
  const int c0 = blockIdx.y * 16;
  const int w  = blockIdx.z;
  const float* W = (w == 0) ? W0 : (w == 1) ? W1 : W2;
  float* Pw = P + (size_t)w * NHH * TT * HSS;

  const float* Arow = x + (size_t)(t0 + lm) * CC;
  v8f acc = {0.f, 0.f, 0.f, 0.f, 0.f, 0.f, 0.f, 0.f};
  for (int k = 0; k < CC; k += 4) {
    v2f a, b;
    a.x = Arow[k + lk];
    a.y = Arow[k + lk + 1];
    b.x = W[(size_t)(k + lk) * CC + c0 + lm];
    b.y = W[(size_t)(k + lk + 1) * CC + c0 + lm];
    if (k + 16 < CC)
      __builtin_prefetch(&W[(size_t)(k + 16 + lk) * CC + c0 + lm], 0, 1);
    acc = wmma4(a, b, acc);
  }
  // scatter to head-major layout (h, t, d)
  const int h  = c0 >> 6;
  const int d0 = c0 & 63;
  float* Ph = Pw + (size_t)h * TT * HSS;
#pragma unroll
  for (int r = 0; r < 8; ++r)
    Ph[(size_t)(t0 + r + drow) * HSS + d0 + lm] = acc[r];
}

// ---------------------------------------------------------------------------
// Kernel 2: per-head value projection V[h] = P2[h] @ Wv  (K = 64)
// ---------------------------------------------------------------------------
__global__ void vproj_kernel(const float* __restrict__ P2,
                             const float* __restrict__ Wv,
                             float* __restrict__ V) {
  const int lane = threadIdx.x;
  const int lm   = lane & 15;
  const int lk   = (lane >> 4) << 1;
  const int drow = (lane >> 4) << 3;
  const int t0 = blockIdx.x * 16;
  const int n0 = blockIdx.y * 16;
  const int h  = blockIdx.z;
  const float* A = P2 + (size_t)h * TT * HSS;
  float* Vh = V + (size_t)h * TT * HSS;

  v8f acc = {0.f, 0.f, 0.f, 0.f, 0.f, 0.f, 0.f, 0.f};
  for (int k = 0; k < HSS; k += 4) {
    v2f a, b;
    a.x = A[(size_t)(t0 + lm) * HSS + k + lk];
    a.y = A[(size_t)(t0 + lm) * HSS + k + lk + 1];
    b.x = Wv[(size_t)(k + lk) * HSS + n0 + lm];
    b.y = Wv[(size_t)(k + lk + 1) * HSS + n0 + lm];
    acc = wmma4(a, b, acc);
  }
#pragma unroll
  for (int r = 0; r < 8; ++r)
    Vh[(size_t)(t0 + r + drow) * HSS + n0 + lm] = acc[r];
}

// ---------------------------------------------------------------------------
// Kernel 3: Vsuf[h][i][d] = sum_{j>i} V[h][j][d]  (exclusive suffix sum)
// ---------------------------------------------------------------------------
__global__ void vsuffix_kernel(const float* __restrict__ V,
                               float* __restrict__ VS) {
  const int h = blockIdx.x;
  const int d = threadIdx.x;  // 64 threads
  const float* Vh = V + (size_t)h * TT * HSS;
  float* Sh = VS + (size_t)h * TT * HSS;
  float run = 0.0f;
  for (int j = TT - 1; j >= 0; --j) {
    Sh[(size_t)j * HSS + d] = run;
    run += Vh[(size_t)j * HSS + d];
  }
}

// ---------------------------------------------------------------------------
// Kernel 4: ATT[h][i][j] = softmax_j(causal(P0[h] @ P1[h]^T * SCALE))
// One wave per (head, 16-row strip). A operand cached in 32 VGPRs across all
// j-tiles (reuse); 16x512 logit strip staged in LDS (32 KB) for the softmax.
// Only j <= i entries are written (rest are exactly zero and handled in
// closed form by the order-3 kernel).
// ---------------------------------------------------------------------------
__global__ void att_softmax_kernel(const float* __restrict__ P0,
                                   const float* __restrict__ P1,
                                   float* __restrict__ ATT) {
  __shared__ float lgt[16 * TT];  // 32 KB logit strip
  const int lane = threadIdx.x;
  const int lm   = lane & 15;
  const int lk   = (lane >> 4) << 1;
  const int drow = (lane >> 4) << 3;
  const int i0 = blockIdx.x * 16;
  const int h  = blockIdx.y;
  const float* P0h = P0 + (size_t)h * TT * HSS;
  const float* P1h = P1 + (size_t)h * TT * HSS;

  // Preload A (16 rows x K=64) into registers: 16 x v2f = 32 VGPRs.
  v2f areg[16];
#pragma unroll
  for (int kk = 0; kk < 16; ++kk) {
    const int k = kk * 4 + lk;
    areg[kk].x = P0h[(size_t)(i0 + lm) * HSS + k];
    areg[kk].y = P0h[(size_t)(i0 + lm) * HSS + k + 1];
  }

  const int jtmax = (i0 + 15) >> 4;  // last j-tile touching the causal region
  for (int jt = 0; jt <= jtmax; ++jt) {
    const int j0 = jt * 16;
    v8f acc = {0.f, 0.f, 0.f, 0.f, 0.f, 0.f, 0.f, 0.f};
#pragma unroll
    for (int kk = 0; kk < 16; ++kk) {
      const int k = kk * 4 + lk;
      v2f b;  // B = P1^T : column n of B is row (j0+n) of P1
      b.x = P1h[(size_t)(j0 + lm) * HSS + k];
      b.y = P1h[(size_t)(j0 + lm) * HSS + k + 1];
      acc = wmma4(areg[kk], b, acc);
    }
#pragma unroll
    for (int r = 0; r < 8; ++r) {
      const int row = r + drow;
      const int j = j0 + lm;
      float val = acc[r] * SCALE;
      if (j > i0 + row) val = -3.0e38f;  // causal mask (never read anyway)
      lgt[row * TT + j] = val;
    }
  }
  __syncthreads();

  // Row softmax: half-wave pair (lane, lane^16) shares one row.
  const int row  = lane & 15;
  const int half = lane >> 4;
  const int len  = i0 + row + 1;  // valid columns 0..len-1
  float mr = -3.0e38f;
  for (int j = half; j < len; j += 2) mr = fmaxf(mr, lgt[row * TT + j]);
  mr = fmaxf(mr, __shfl_xor(mr, 16));
  float sr = 0.0f;
  for (int j = half; j < len; j += 2) sr += __expf(lgt[row * TT + j] - mr);
  sr += __shfl_xor(sr, 16);
  const float inv = 1.0f / sr;
  float* arow = ATT + ((size_t)h * TT + (i0 + row)) * TT;
  for (int j = half; j < len; j += 2)
    arow[j] = __expf(lgt[row * TT + j] - mr) * inv;
}

// ---------------------------------------------------------------------------
// Kernel 5: order-3 step, single streaming pass with online softmax.
// For each (h, i): s[j,d] = att[i,j] * p2[j,d] over j<=i; the masked region
// (j>i) is exactly zero, handled in closed form:
//   denom += (T-1-i)*HS * exp(-M)
//   y[d]  += exp(-M) * Vsuf[i,d]
// One 256-thread block per (h,i): thread = (j-group g in 0..3, d in 0..63).
// ---------------------------------------------------------------------------
__global__ void order3_kernel(const float* __restrict__ ATT,
                              const float* __restrict__ P2,
                              const float* __restrict__ V,
                              const float* __restrict__ VSUF,
                              float* __restrict__ Y) {
  __shared__ float s_att[TT];
  __shared__ float s_red[256];
  __shared__ float s_acc[256];
  const int i = blockIdx.x;
  const int h = blockIdx.y;
  const int t = threadIdx.x;
  const int d = t & 63;
  const int g = t >> 6;

  const float* attrow = ATT + ((size_t)h * TT + i) * TT;
  const float* P2h = P2 + (size_t)h * TT * HSS;
  const float* Vh  = V  + (size_t)h * TT * HSS;

  for (int j = t; j <= i; j += 256) s_att[j] = attrow[j];
  __syncthreads();

  // Online (rescaling) max / denominator / weighted accumulator per thread.
  float m = -3.0e38f, dsum = 0.0f, acc = 0.0f;
  for (int j = g; j <= i; j += 4) {
    const float s  = s_att[j] * P2h[(size_t)j * HSS + d];
    const float vv = Vh[(size_t)j * HSS + d];
    if (s <= m) {
      const float e = __expf(s - m);
      dsum += e;
      acc += e * vv;
    } else {
      const float sc = __expf(m - s);
      dsum = dsum * sc + 1.0f;
      acc  = acc * sc + vv;
      m = s;
    }
  }

  // Block max of m; include 0 (value of all masked entries) — shift-safe.
  s_red[t] = m;
  __syncthreads();
  for (int off = 128; off > 0; off >>= 1) {
    if (t < off) s_red[t] = fmaxf(s_red[t], s_red[t + off]);
    __syncthreads();
  }
  const float M = fmaxf(s_red[0], 0.0f);
  __syncthreads();

  const float sc = __expf(m - M);  // underflows to 0 for idle threads
  s_red[t] = dsum * sc;
  s_acc[t] = acc * sc;
  __syncthreads();
  for (int off = 128; off > 0; off >>= 1) {
    if (t < off) s_red[t] += s_red[t + off];
    __syncthreads();
  }
  const float e0 = __expf(-M);
  const float denom = s_red[0] + (float)((TT - 1 - i) * HSS) * e0;
  if (t < 64) {
    const float accT = s_acc[d] + s_acc[64 + d] + s_acc[128 + d] + s_acc[192 + d];
    const float vs = VSUF[(size_t)h * TT * HSS + (size_t)i * HSS + d];
    Y[(size_t)i * CC + h * HSS + d] = (accT + e0 * vs) / denom;
  }
}

// ---------------------------------------------------------------------------
// Kernel 6: out = Y @ Wc  (512x768 @ 768x768), WMMA f32 like kernel 1.
// ---------------------------------------------------------------------------
__global__ void out_proj_kernel(const float* __restrict__ Y,
                                const float* __restrict__ Wc,
                                float* __restrict__ out) {
  const int lane = threadIdx.x;
  const int lm   = lane & 15;
  const int lk   = (lane >> 4) << 1;
  const int drow = (lane >> 4) << 3;
  const int t0 = blockIdx.x * 16;
  const int c0 = blockIdx.y * 16;

  const float* Arow = Y + (size_t)(t0 + lm) * CC;
  v8f acc = {0.f, 0.f, 0.f, 0.f, 0.f, 0.f, 0.f, 0.f};
  for (int k = 0; k < CC; k += 4) {
    v2f a, b;
    a.x = Arow[k + lk];
    a.y = Arow[k + lk + 1];
    b.x = Wc[(size_t)(k + lk) * CC + c0 + lm];
    b.y = Wc[(size_t)(k + lk + 1) * CC + c0 + lm];
    if (k + 16 < CC)
      __builtin_prefetch(&Wc[(size_t)(k + 16 + lk) * CC + c0 + lm], 0, 1);
    acc = wmma4(a, b, acc);
  }
#pragma unroll
  for (int r = 0; r < 8; ++r)
    out[(size_t)(t0 + r + drow) * CC + c0 + lm] = acc[r];
}

// ---------------------------------------------------------------------------
// Launch pipeline
// ---------------------------------------------------------------------------
extern "C" void kernel_launch(void* const* d_in, const int* in_sizes, int n_in,
                              void* d_out, int out_size, void* d_ws,
                              size_t ws_size, hipStream_t stream) {
  (void)in_sizes; (void)n_in; (void)out_size; (void)ws_size;
  const float* x  = (const float*)d_in[0];
  const float* W0 = (const float*)d_in[1];
  const float* W1 = (const float*)d_in[2];
  const float* W2 = (const float*)d_in[3];
  const float* Wv = (const float*)d_in[4];
  const float* Wc = (const float*)d_in[5];
  float* out = (float*)d_out;
  float* ws  = (float*)d_ws;

  const size_t NPH = (size_t)NHH * TT * HSS;  // 393216 floats per projection
  float* P0  = ws;                            // [NH][T][HS]
  float* P1  = ws + NPH;
  float* P2  = ws + 2 * NPH;
  float* V   = ws + 3 * NPH;
  float* VS  = ws + 4 * NPH;
  float* ATT = ws + 5 * NPH;                  // [NH][T][T] = 12.6 MB
  float* Y   = ATT + (size_t)NHH * TT * TT;   // [T][C]
  // total workspace: ~22 MB (L2-resident on MI455X, 192 MB L2)

  qkv_proj_kernel<<<dim3(TT / 16, CC / 16, 3), 32, 0, stream>>>(x, W0, W1, W2, P0);
  vproj_kernel<<<dim3(TT / 16, HSS / 16, NHH), 32, 0, stream>>>(P2, Wv, V);
  vsuffix_kernel<<<dim3(NHH), 64, 0, stream>>>(V, VS);
  att_softmax_kernel<<<dim3(TT / 16, NHH), 32, 0, stream>>>(P0, P1, ATT);
  order3_kernel<<<dim3(TT, NHH), 256, 0, stream>>>(ATT, P2, V, VS, Y);
  out_proj_kernel<<<dim3(TT / 16, CC / 16), 32, 0, stream>>>(Y, Wc, out);
}